// DynamicGroupEncoder_22866405884231
// MI455X (gfx1250) — compile-verified
//
#include <hip/hip_runtime.h>

typedef __attribute__((ext_vector_type(2))) float v2f;
typedef __attribute__((ext_vector_type(8))) float v8f;

#define HIST    200
#define EMB     64
#define EMB_PAD 68      // floats per LDS row = 272 B: 16B-aligned for B128, conflict-friendly
#define NEG_INF (-1e9f)

__global__ __launch_bounds__(256)
void dge_kernel(const float* __restrict__ item_emb,
                const int*   __restrict__ history_ids,
                const int*   __restrict__ history_mask,
                const float* __restrict__ attn_w,
                const float* __restrict__ attn_b,
                float*       __restrict__ out)
{
    __shared__ float s_emb[HIST * EMB_PAD];   // 54,400 B  gathered embeddings
    __shared__ float s_attn[HIST];            // softmax weights
    __shared__ float s_w[EMB];                // attn_w broadcast
    __shared__ float s_partial[8 * EMB];      // per-wave WMMA partials
    __shared__ float s_red[32];               // block reductions
    __shared__ int   s_id[HIST];
    __shared__ int   s_m[HIST];

    const int tid  = threadIdx.x;
    const int b    = blockIdx.x;
    const int wave = tid >> 5;
    const int lane = tid & 31;

    // ---------------- Phase A: stage ids / mask / attn_w ----------------
    if (tid < HIST) {
        s_id[tid] = history_ids[(size_t)b * HIST + tid];
        s_m[tid]  = history_mask[(size_t)b * HIST + tid];
    }
    if (tid < EMB) s_w[tid] = attn_w[tid];
    __syncthreads();

    // ---------------- Phase B: async gather rows -> LDS ----------------
    // 16 lanes per row, 16 B per lane. Masked rows are zero-filled (their
    // softmax weight is exactly 0; zeros avoid NaN poisoning the WMMA).
    {
        const int c  = tid & 15;   // 16B chunk within a 256B row
        const int r0 = tid >> 4;   // 16 rows per pass
        const unsigned long long gbase = (unsigned long long)(const void*)item_emb;
        #pragma unroll
        for (int p = 0; p < (HIST + 15) / 16; ++p) {
            int h = p * 16 + r0;
            if (h < HIST) {
                // low 32 bits of a generic pointer to __shared__ == LDS byte offset
                unsigned lds_off = (unsigned)(unsigned long long)(&s_emb[h * EMB_PAD + c * 4]);
                if (s_m[h] > 0) {
                    unsigned long long ga = gbase
                        + (unsigned long long)(unsigned)s_id[h] * (EMB * 4ull)
                        + (unsigned long long)(c * 16);
                    asm volatile("global_load_async_to_lds_b128 %0, %1, off"
                                 :: "v"(lds_off), "v"(ga) : "memory");
                } else {
                    *(float4*)&s_emb[h * EMB_PAD + c * 4] = make_float4(0.f, 0.f, 0.f, 0.f);
                }
            }
        }
        asm volatile("s_wait_asynccnt 0x0" ::: "memory");
    }
    __syncthreads();

    // ---------------- Phase C: masked logits ----------------
    float logit = NEG_INF;
    if (tid < HIST) {
        float acc = 0.f;
        #pragma unroll
        for (int e = 0; e < EMB; e += 4) {
            float4 ev = *(const float4*)&s_emb[tid * EMB_PAD + e];
            float4 wv = *(const float4*)&s_w[e];     // same addr across lanes -> broadcast
            acc = fmaf(ev.x, wv.x, acc);
            acc = fmaf(ev.y, wv.y, acc);
            acc = fmaf(ev.z, wv.z, acc);
            acc = fmaf(ev.w, wv.w, acc);
        }
        logit = (s_m[tid] > 0) ? (acc + attn_b[0]) : NEG_INF;
    }

    // ---------------- Phase D: softmax over HIST ----------------
    // max-reduce (extra threads hold NEG_INF, so they are no-ops)
    float v = logit;
    #pragma unroll
    for (int o = 16; o > 0; o >>= 1) v = fmaxf(v, __shfl_xor(v, o, 32));
    if (lane == 0) s_red[wave] = v;
    __syncthreads();
    if (tid < 8) {
        float x = s_red[tid];
        #pragma unroll
        for (int o = 4; o > 0; o >>= 1) x = fmaxf(x, __shfl_xor(x, o, 8));
        if (tid == 0) s_red[16] = x;
    }
    __syncthreads();
    const float mx = s_red[16];

    float ex = (tid < HIST) ? __expf(logit - mx) : 0.f;   // masked: exp(-1e9-mx) -> 0
    float sv = ex;
    #pragma unroll
    for (int o = 16; o > 0; o >>= 1) sv += __shfl_xor(sv, o, 32);
    __syncthreads();                    // s_red reuse
    if (lane == 0) s_red[wave] = sv;
    __syncthreads();
    if (tid < 8) {
        float x = s_red[tid];
        #pragma unroll
        for (int o = 4; o > 0; o >>= 1) x += __shfl_xor(x, o, 8);
        if (tid == 0) s_red[17] = x;
    }
    __syncthreads();
    const float inv = 1.0f / s_red[17];
    if (tid < HIST) s_attn[tid] = ex * inv;
    __syncthreads();

    // ---------------- Phase E: weighted sum via V_WMMA_F32_16X16X4_F32 ----------------
    // D[m][n] = sum_k emb[h0+k][e0+m] * attn[h0+k]  (B broadcast along n),
    // so every column of D is the partial output for e0+m. 8 waves split the
    // 50 k-tiles; e covered by 4 tiles of 16.
    // f32 16x4 A layout: lane L, vgpr v -> A[M = L%16][K = 2*(L/16) + v]
    // f32 4x16 B layout: lane L, vgpr v -> B[K = 2*(L/16) + v][N = L%16]
    #pragma unroll
    for (int n = 0; n < 4; ++n) {
        v8f c = {};
        for (int k4 = wave; k4 < HIST / 4; k4 += 8) {   // wave-uniform trip count
            const int h = 4 * k4 + 2 * (lane >> 4);
            const int e = 16 * n + (lane & 15);
            v2f a, bm;
            a[0]  = s_emb[(h + 0) * EMB_PAD + e];
            a[1]  = s_emb[(h + 1) * EMB_PAD + e];
            bm[0] = s_attn[h + 0];
            bm[1] = s_attn[h + 1];
            c = __builtin_amdgcn_wmma_f32_16x16x4_f32(
                    false, a, false, bm, (short)0, c, false, false);
        }
        // D layout: lanes 0-15 vgpr r -> M=r ; lanes 16-31 vgpr r -> M=8+r.
        // Columns identical -> only lanes with N==0 need to write.
        if ((lane & 15) == 0) {
            const int base = wave * EMB + 16 * n + ((lane >> 4) << 3);
            #pragma unroll
            for (int r = 0; r < 8; ++r) s_partial[base + r] = c[r];
        }
    }
    __syncthreads();

    // ---------------- Phase F: cross-wave reduce + store ----------------
    if (tid < EMB) {
        float acc = 0.f;
        #pragma unroll
        for (int w = 0; w < 8; ++w) acc += s_partial[w * EMB + tid];
        out[(size_t)b * EMB + tid] = acc;
    }
}

extern "C" void kernel_launch(void* const* d_in, const int* in_sizes, int n_in,
                              void* d_out, int out_size, void* d_ws, size_t ws_size,
                              hipStream_t stream) {
    const float* item_emb = (const float*)d_in[0];
    const int*   ids      = (const int*)  d_in[1];
    const int*   mask     = (const int*)  d_in[2];
    const float* w        = (const float*)d_in[3];
    const float* bia      = (const float*)d_in[4];
    float*       out      = (float*)d_out;

    const int B = in_sizes[1] / HIST;   // 16384
    dge_kernel<<<B, 256, 0, stream>>>(item_emb, ids, mask, w, bia, out);
}